// MultiScaleRetention_25314537243329
// MI455X (gfx1250) — compile-verified
//
#include <hip/hip_runtime.h>
#include <hip/hip_bf16.h>

#define BB 4
#define SS 1024
#define HIDD 1024
#define HH 8
#define HDD 128
#define EPSV 1e-5f

typedef float v2f __attribute__((ext_vector_type(2)));
typedef float v8f __attribute__((ext_vector_type(8)));
typedef unsigned int u32x4 __attribute__((ext_vector_type(4)));
typedef int i32x4 __attribute__((ext_vector_type(4)));
typedef int i32x8 __attribute__((ext_vector_type(8)));

// D = A(16x4) * B(4x16) + C, fp32 WMMA (CDNA5 V_WMMA_F32_16X16X4_F32)
__device__ __forceinline__ v8f wmma4(v2f a, v2f b, v8f c) {
  return __builtin_amdgcn_wmma_f32_16x16x4_f32(
      /*neg_a=*/false, a, /*neg_b=*/false, b,
      /*c_mod=*/(short)0, c, /*reuse_a=*/false, /*reuse_b=*/false);
}

// A fragment from row-major [rows][ld] LDS tile (also serves as B^T loader).
__device__ __forceinline__ v2f ldA(const float* s, int ld, int rbase, int k0,
                                   int lmod, int lh) {
  const float* p = s + (size_t)(rbase + lmod) * ld + k0 + 2 * lh;
  v2f a;
  a.x = p[0];
  a.y = p[1];
  return a;
}

// B fragment from row-major [K][cols] LDS tile: b[v] = B[k0+2*lh+v][cbase+lmod]
__device__ __forceinline__ v2f ldB(const float* s, int ld, int cbase, int k0,
                                   int lmod, int lh) {
  v2f b;
  b.x = s[(size_t)(k0 + 2 * lh + 0) * ld + cbase + lmod];
  b.y = s[(size_t)(k0 + 2 * lh + 1) * ld + cbase + lmod];
  return b;
}

// ---------------------------------------------------------------------------
// Tensor Data Mover: async 2D f32 tile (rows x width, row stride in elements)
// from global to LDS byte offset. D# per CDNA5 ISA ch.8 (group0/group1).
// Issue from ONE wave only (TDM ignores EXEC): guard with a wave-level branch.
// ---------------------------------------------------------------------------
__device__ __forceinline__ void tdm_load_2d(unsigned lds_byte_off,
                                            const void* gptr, int width,
                                            int rows, int stride_elems) {
  unsigned long long ga = (unsigned long long)gptr;
  u32x4 g0;
  g0[0] = 1u;                                   // count=1, user D#
  g0[1] = lds_byte_off;                         // lds_addr
  g0[2] = (unsigned)(ga & 0xffffffffu);         // global_addr[31:0]
  g0[3] = (unsigned)((ga >> 32) & 0x01ffffffu)  // global_addr[56:32]
          | (2u << 30);                         // type=2 (image)
  i32x8 g1;
  g1[0] = 2 << 16;                                  // data_size = 4B
  g1[1] = (width & 0xffff) << 16;                   // tensor_dim0 lo
  g1[2] = (width >> 16) | ((rows & 0xffff) << 16);  // dim0 hi | dim1 lo
  g1[3] = (rows >> 16) | ((width & 0xffff) << 16);  // dim1 hi | tile_dim0
  g1[4] = rows & 0xffff;                            // tile_dim1 (tile_dim2=0)
  g1[5] = stride_elems;                             // tensor_dim0_stride lo
  g1[6] = 0;
  g1[7] = 0;
  i32x4 z4 = {0, 0, 0, 0};
#if __clang_major__ >= 23
  i32x8 z8 = {0, 0, 0, 0, 0, 0, 0, 0};
  __builtin_amdgcn_tensor_load_to_lds(g0, g1, z4, z4, z8, 0);
#else
  __builtin_amdgcn_tensor_load_to_lds(g0, g1, z4, z4, 0);
#endif
}

__device__ __forceinline__ void tdm_wait0() {
  __builtin_amdgcn_s_wait_tensorcnt(0);
}

// ---------------------------------------------------------------------------
// Kernel 1: per-head QKV projection + Theta phase split.
// grid = B*H*(S/64), block = 256 (8 waves), dyn LDS = (64*128 + 128*128)*4
// ---------------------------------------------------------------------------
__global__ __launch_bounds__(256) void qkv_kernel(
    const float* __restrict__ X, const float* __restrict__ WQ,
    const float* __restrict__ WK, const float* __restrict__ WV,
    const float* __restrict__ theta, float* __restrict__ Qc,
    float* __restrict__ Qs, float* __restrict__ Kc, float* __restrict__ Ks,
    float* __restrict__ Vp) {
  extern __shared__ float sm[];
  float* Xs = sm;             // 64 x 128  (LDS byte 0)
  float* Ws = sm + 64 * 128;  // 128 x 128 (LDS byte 32768)

  const int tid = threadIdx.x;
  const int blk = blockIdx.x;
  const int t = blk & 15;
  const int bh = blk >> 4;
  const int h = bh & (HH - 1);
  const int bat = bh >> 3;
  const int n0 = t * 64;

  // TDM: X tile (64 rows x 128 cols of head h, row stride HID)
  if (tid < 32) {
    tdm_load_2d(0, X + ((size_t)(bat * SS + n0) * HIDD) + h * HDD, 128, 64,
                HIDD);
  }

  const int lane = tid & 31, wave = tid >> 5;
  const int lmod = lane & 15, lh = lane >> 4;
  const int ni = wave >> 1;          // n-subtile 0..3
  const int dbase = (wave & 1) * 4;  // 4 d-subtiles per wave

  for (int wsel = 0; wsel < 3; ++wsel) {
    const float* W =
        (wsel == 0 ? WQ : (wsel == 1 ? WK : WV)) + (size_t)h * HDD * HDD;
    if (tid < 32) {
      tdm_load_2d(64 * 128 * 4, W, 128, 128, 128);  // weight tile 128x128
      tdm_wait0();  // in-order with X-tile load on first iteration
    }
    __syncthreads();

    v8f acc[4] = {};
    for (int k = 0; k < 128; k += 4) {
      v2f a = ldA(Xs, 128, ni * 16, k, lmod, lh);  // reused across 4 tiles
#pragma unroll
      for (int j = 0; j < 4; ++j) {
        v2f b = ldB(Ws, 128, (dbase + j) * 16, k, lmod, lh);
        acc[j] = wmma4(a, b, acc[j]);
      }
    }
#pragma unroll
    for (int j = 0; j < 4; ++j) {
      const int cb = (dbase + j) * 16;
#pragma unroll
      for (int r = 0; r < 8; ++r) {
        int row = ni * 16 + r + 8 * lh;
        int sg = n0 + row;
        int dc = cb + lmod;
        size_t o = ((size_t)(bat * HH + h) * SS + sg) * HDD + dc;
        float val = acc[j][r];
        if (wsel == 2) {
          Vp[o] = val;
        } else {
          float ph = (float)(sg + 1) * theta[h * HDD + dc];
          float cs = __cosf(ph), sn = __sinf(ph);
          if (wsel == 0) {
            Qc[o] = val * cs;
            Qs[o] = val * sn;
          } else {
            Kc[o] = val * cs;
            Ks[o] = val * sn;
          }
        }
      }
    }
    __syncthreads();
  }
}

// ---------------------------------------------------------------------------
// Kernel 2: flash-style causal retention with per-head decay.
// grid = B*H*(S/64), block = 256 (8 waves), dyn LDS = 128KB
// ---------------------------------------------------------------------------
__global__ __launch_bounds__(256) void attn_kernel(
    const float* __restrict__ Qc, const float* __restrict__ Qs,
    const float* __restrict__ Kc, const float* __restrict__ Ks,
    const float* __restrict__ Vp, float* __restrict__ Yre,
    float* __restrict__ Yim) {
  extern __shared__ float sm[];
  float* Qcs = sm;          // 64x128  byte 0
  float* Qss = Qcs + 8192;  // 64x128  byte 32768
  float* Kcs = Qss + 8192;  // 32x128  byte 65536
  float* Kss = Kcs + 4096;  // 32x128  byte 81920
  float* Vs = Kss + 4096;   // 32x128  byte 98304
  float* Are = Vs + 4096;   // 64x32
  float* Aim = Are + 2048;  // 64x32

  const int tid = threadIdx.x;
  const int blk = blockIdx.x;
  const int t = blk & 15;
  const int bh = blk >> 4;
  const int h = bh & (HH - 1);
  const int n0 = t * 64;

  const float l0 = -3.4657359028f;  // log(1/32)
  const float l1 = -6.2383246250f;  // log(1/512)
  float gl = l0 + (l1 - l0) * ((float)h / (float)(HH - 1));
  float gam = 1.0f - __expf(gl);
  float lg = __logf(gam);

  const size_t qbase = ((size_t)bh * SS + n0) * HDD;

  // TDM: stage Q (re/im) tiles once (contiguous rows, stride 128)
  if (tid < 32) {
    tdm_load_2d(0, Qc + qbase, 128, 64, 128);
    tdm_load_2d(32768, Qs + qbase, 128, 64, 128);
  }

  const int lane = tid & 31, wave = tid >> 5;
  const int lmod = lane & 15, lh = lane >> 4;
  const int ni = wave >> 1, mj = wave & 1;   // stage-1 roles
  const int ni2 = wave & 3, dh = wave >> 2;  // stage-2 roles

  v8f yre[4] = {}, yim[4] = {};

  const int mblocks = (n0 >> 5) + 2;  // causal: m <= n0+63
  for (int mb = 0; mb < mblocks; ++mb) {
    const int m0 = mb * 32;
    if (tid < 32) {  // TDM: K(re/im), V m-block; waits cover Q loads too
      const size_t kb = ((size_t)bh * SS + m0) * HDD;
      tdm_load_2d(65536, Kc + kb, 128, 32, 128);
      tdm_load_2d(81920, Ks + kb, 128, 32, 128);
      tdm_load_2d(98304, Vp + kb, 128, 32, 128);
      tdm_wait0();
    }
    __syncthreads();

    // stage 1: complex score tile 16x16 per wave over full d=128
    {
      v8f are = {}, aim = {};
      for (int k = 0; k < 128; k += 4) {
        v2f ac = ldA(Qcs, 128, ni * 16, k, lmod, lh);
        v2f as_ = ldA(Qss, 128, ni * 16, k, lmod, lh);
        v2f bc = ldA(Kcs, 128, mj * 16, k, lmod, lh);  // K^T as B operand
        v2f bs = ldA(Kss, 128, mj * 16, k, lmod, lh);
        are = wmma4(ac, bc, are);
        are = wmma4(as_, bs, are);
        aim = wmma4(as_, bc, aim);
        v2f acn = -ac;  // f32 WMMA has no A-negate; negate in VALU
        aim = wmma4(acn, bs, aim);
      }
#pragma unroll
      for (int r = 0; r < 8; ++r) {
        int n = n0 + ni * 16 + r + 8 * lh;
        int m = m0 + mj * 16 + lmod;
        float w = (n >= m) ? __expf((float)(n - m) * lg) : 0.0f;
        int o = (ni * 16 + r + 8 * lh) * 32 + mj * 16 + lmod;
        Are[o] = are[r] * w;
        Aim[o] = aim[r] * w;
      }
    }
    __syncthreads();

    // stage 2: Y += att * V  (K = 32), A-frags reused across 4 d-tiles
#pragma unroll
    for (int kk = 0; kk < 32; kk += 4) {
      v2f ar = ldA(Are, 32, ni2 * 16, kk, lmod, lh);
      v2f ai = ldA(Aim, 32, ni2 * 16, kk, lmod, lh);
#pragma unroll
      for (int j = 0; j < 4; ++j) {
        v2f bv = ldB(Vs, 128, dh * 64 + j * 16, kk, lmod, lh);
        yre[j] = wmma4(ar, bv, yre[j]);
        yim[j] = wmma4(ai, bv, yim[j]);
      }
    }
    __syncthreads();
  }

#pragma unroll
  for (int j = 0; j < 4; ++j) {
#pragma unroll
    for (int r = 0; r < 8; ++r) {
      int n = n0 + ni2 * 16 + r + 8 * lh;
      int dc = dh * 64 + j * 16 + lmod;
      size_t o = ((size_t)bh * SS + n) * HDD + dc;
      Yre[o] = yre[j][r];
      Yim[o] = yim[j][r];
    }
  }
}

// ---------------------------------------------------------------------------
// Kernel 3: G = X @ W_G (complex weights), fused complex swish.
// grid = 1024, block = 256, dyn LDS = 48KB
// ---------------------------------------------------------------------------
__global__ __launch_bounds__(256) void gate_kernel(
    const float* __restrict__ X, const float* __restrict__ WG,  // interleaved
    float* __restrict__ Sre, float* __restrict__ Sim) {
  extern __shared__ float sm[];
  float* As = sm;           // 64x64  byte 0
  float* Bre = As + 4096;   // 64x64  byte 16384
  float* Bim = Bre + 4096;  // 64x64  byte 32768

  const int tid = threadIdx.x;
  const int rt = blockIdx.x >> 4, ct = blockIdx.x & 15;
  const int r0 = rt * 64, c0 = ct * 64;

  const int lane = tid & 31, wave = tid >> 5;
  const int lmod = lane & 15, lh = lane >> 4;
  const int ni = wave >> 1, cjb = (wave & 1) * 2;

  v8f accr[2] = {}, acci[2] = {};

  for (int kb = 0; kb < 16; ++kb) {
    const int k0 = kb * 64;
    if (tid < 32) {  // TDM A tile overlaps cooperative B loads below
      tdm_load_2d(0, X + (size_t)r0 * HIDD + k0, 64, 64, HIDD);
    }
    for (int idx = tid; idx < 2048; idx += 256) {  // B: 64x64 complex
      int kr = idx >> 5, cc2 = idx & 31;           // float4 = 2 complex
      float4 v =
          ((const float4*)(WG + ((size_t)(k0 + kr) * HIDD + c0) * 2))[cc2];
      Bre[kr * 64 + cc2 * 2 + 0] = v.x;
      Bim[kr * 64 + cc2 * 2 + 0] = v.y;
      Bre[kr * 64 + cc2 * 2 + 1] = v.z;
      Bim[kr * 64 + cc2 * 2 + 1] = v.w;
    }
    if (tid < 32) tdm_wait0();
    __syncthreads();
    for (int k = 0; k < 64; k += 4) {
      v2f a = ldA(As, 64, ni * 16, k, lmod, lh);  // reused for re & im tiles
#pragma unroll
      for (int j = 0; j < 2; ++j) {
        const int cb = (cjb + j) * 16;
        v2f br = ldB(Bre, 64, cb, k, lmod, lh);
        v2f bi = ldB(Bim, 64, cb, k, lmod, lh);
        accr[j] = wmma4(a, br, accr[j]);
        acci[j] = wmma4(a, bi, acci[j]);
      }
    }
    __syncthreads();
  }

#pragma unroll
  for (int j = 0; j < 2; ++j) {
#pragma unroll
    for (int r = 0; r < 8; ++r) {
      int tok = r0 + ni * 16 + r + 8 * lh;
      int col = c0 + (cjb + j) * 16 + lmod;
      float gr = accr[j][r], gi = acci[j][r];
      // complex swish: G * 1/(1+exp(-G))
      float e = __expf(-gr);
      float zr = 1.0f + e * __cosf(gi);
      float zi = -e * __sinf(gi);
      float den = zr * zr + zi * zi;
      float sr = (gr * zr + gi * zi) / den;
      float si = (gi * zr - gr * zi) / den;
      size_t o = (size_t)tok * HIDD + col;
      Sre[o] = sr;
      Sim[o] = si;
    }
  }
}

// ---------------------------------------------------------------------------
// Kernel 4: complex group-norm (groups = heads, unbiased var) * swish gate.
// ---------------------------------------------------------------------------
__global__ __launch_bounds__(256) void gnorm_kernel(
    const float* __restrict__ Yre, const float* __restrict__ Yim,
    const float* __restrict__ Sre, const float* __restrict__ Sim,
    const float* __restrict__ gnw, const float* __restrict__ gnb,
    float* __restrict__ Gr, float* __restrict__ Gi) {
  const int tok = blockIdx.x;
  const int bat = tok / SS, s = tok % SS;
  const int h = threadIdx.x >> 5, lane = threadIdx.x & 31;
  const size_t base = ((size_t)(bat * HH + h) * SS + s) * HDD;

  float yr[4], yi[4];
  float sr_ = 0.f, si_ = 0.f;
#pragma unroll
  for (int j = 0; j < 4; ++j) {
    int d = lane + 32 * j;
    yr[j] = Yre[base + d];
    yi[j] = Yim[base + d];
    sr_ += yr[j];
    si_ += yi[j];
  }
#pragma unroll
  for (int m = 16; m >= 1; m >>= 1) {
    sr_ += __shfl_xor(sr_, m, 32);
    si_ += __shfl_xor(si_, m, 32);
  }
  float mr = sr_ * (1.0f / HDD), mi = si_ * (1.0f / HDD);
  float sv = 0.f;
#pragma unroll
  for (int j = 0; j < 4; ++j) {
    float dr = yr[j] - mr, di = yi[j] - mi;
    sv += dr * dr + di * di;
  }
#pragma unroll
  for (int m = 16; m >= 1; m >>= 1) sv += __shfl_xor(sv, m, 32);
  float inv = rsqrtf(sv * (1.0f / (HDD - 1)) + EPSV);

#pragma unroll
  for (int j = 0; j < 4; ++j) {
    int d = lane + 32 * j;
    int col = h * HDD + d;
    float w = gnw[col], bi_ = gnb[col];
    float ynr = (yr[j] - mr) * inv * w + bi_;
    float yni = (yi[j] - mi) * inv * w;
    size_t o = (size_t)tok * HIDD + col;
    float sr = Sre[o], si = Sim[o];
    Gr[o] = sr * ynr - si * yni;  // swish * Yn (complex)
    Gi[o] = sr * yni + si * ynr;
  }
}

// ---------------------------------------------------------------------------
// Kernel 5: Out = Gated @ W_O (complex x complex), interleaved complex64 out.
// grid = 1024, block = 256, dyn LDS = 64KB
// ---------------------------------------------------------------------------
__global__ __launch_bounds__(256) void outproj_kernel(
    const float* __restrict__ Gr, const float* __restrict__ Gi,
    const float* __restrict__ WO, float* __restrict__ out) {
  extern __shared__ float sm[];
  float* Ar = sm;           // 64x64  byte 0
  float* Ai = Ar + 4096;    // 64x64  byte 16384
  float* Bre = Ai + 4096;   // 64x64  byte 32768
  float* Bim = Bre + 4096;  // 64x64  byte 49152

  const int tid = threadIdx.x;
  const int rt = blockIdx.x >> 4, ct = blockIdx.x & 15;
  const int r0 = rt * 64, c0 = ct * 64;

  const int lane = tid & 31, wave = tid >> 5;
  const int lmod = lane & 15, lh = lane >> 4;
  const int ni = wave >> 1, cjb = (wave & 1) * 2;

  v8f accr[2] = {}, acci[2] = {};

  for (int kb = 0; kb < 16; ++kb) {
    const int k0 = kb * 64;
    if (tid < 32) {  // TDM A tiles (re, im), overlapped with B loads
      tdm_load_2d(0, Gr + (size_t)r0 * HIDD + k0, 64, 64, HIDD);
      tdm_load_2d(16384, Gi + (size_t)r0 * HIDD + k0, 64, 64, HIDD);
    }
    for (int idx = tid; idx < 2048; idx += 256) {  // B: 64x64 complex
      int kr = idx >> 5, cc2 = idx & 31;
      float4 v =
          ((const float4*)(WO + ((size_t)(k0 + kr) * HIDD + c0) * 2))[cc2];
      Bre[kr * 64 + cc2 * 2 + 0] = v.x;
      Bim[kr * 64 + cc2 * 2 + 0] = v.y;
      Bre[kr * 64 + cc2 * 2 + 1] = v.z;
      Bim[kr * 64 + cc2 * 2 + 1] = v.w;
    }
    if (tid < 32) tdm_wait0();
    __syncthreads();
    for (int k = 0; k < 64; k += 4) {
      v2f ar = ldA(Ar, 64, ni * 16, k, lmod, lh);
      v2f ai = ldA(Ai, 64, ni * 16, k, lmod, lh);
      v2f ain = -ai;
#pragma unroll
      for (int j = 0; j < 2; ++j) {
        const int cb = (cjb + j) * 16;
        v2f br = ldB(Bre, 64, cb, k, lmod, lh);
        v2f bi = ldB(Bim, 64, cb, k, lmod, lh);
        accr[j] = wmma4(ar, br, accr[j]);
        accr[j] = wmma4(ain, bi, accr[j]);  // re -= ai*bi
        acci[j] = wmma4(ar, bi, acci[j]);
        acci[j] = wmma4(ai, br, acci[j]);
      }
    }
    __syncthreads();
  }

#pragma unroll
  for (int j = 0; j < 2; ++j) {
#pragma unroll
    for (int r = 0; r < 8; ++r) {
      int tok = r0 + ni * 16 + r + 8 * lh;
      int col = c0 + (cjb + j) * 16 + lmod;
      size_t o = ((size_t)tok * HIDD + col) * 2;
      out[o + 0] = accr[j][r];
      out[o + 1] = acci[j][r];
    }
  }
}

extern "C" void kernel_launch(void* const* d_in, const int* in_sizes, int n_in,
                              void* d_out, int out_size, void* d_ws,
                              size_t ws_size, hipStream_t stream) {
  const float* X = (const float*)d_in[0];
  const float* WQ = (const float*)d_in[1];
  const float* WK = (const float*)d_in[2];
  const float* WV = (const float*)d_in[3];
  const float* theta = (const float*)d_in[4];
  const float* WG = (const float*)d_in[5];  // complex64 interleaved
  const float* WO = (const float*)d_in[6];  // complex64 interleaved
  const float* gnw = (const float*)d_in[7];
  const float* gnb = (const float*)d_in[8];
  float* out = (float*)d_out;

  const size_t P = (size_t)BB * SS * HIDD;  // 4M floats per plane
  float* ws = (float*)d_ws;
  float* Qc = ws + 0 * P;
  float* Qs = ws + 1 * P;
  float* Kc = ws + 2 * P;
  float* Ks = ws + 3 * P;
  float* Vp = ws + 4 * P;
  float* Yre = ws + 5 * P;
  float* Yim = ws + 6 * P;
  float* Sre = ws + 7 * P;
  float* Sim = ws + 8 * P;
  float* Gr = ws + 9 * P;
  float* Gi = ws + 10 * P;

  const int gQKV = BB * HH * (SS / 64);            // 512
  const int gGEMM = (BB * SS / 64) * (HIDD / 64);  // 1024

  qkv_kernel<<<gQKV, 256, (64 * 128 + 128 * 128) * 4, stream>>>(
      X, WQ, WK, WV, theta, Qc, Qs, Kc, Ks, Vp);
  attn_kernel<<<gQKV, 256, 32768 * 4, stream>>>(Qc, Qs, Kc, Ks, Vp, Yre, Yim);
  gate_kernel<<<gGEMM, 256, 12288 * 4, stream>>>(X, WG, Sre, Sim);
  gnorm_kernel<<<BB * SS, 256, 0, stream>>>(Yre, Yim, Sre, Sim, gnw, gnb, Gr,
                                            Gi);
  outproj_kernel<<<gGEMM, 256, 16384 * 4, stream>>>(Gr, Gi, WO, out);
}